// RC_20332375180115
// MI455X (gfx1250) — compile-verified
//
#include <hip/hip_runtime.h>
#include <hip/hip_bf16.h>
#include <math.h>

typedef __attribute__((ext_vector_type(2))) float v2f;
typedef __attribute__((ext_vector_type(8))) float v8f;

#define B_  64
#define T_  200
#define D_  700
#define H_  1024
#define O_  20
#define M_  (B_*T_)          // 12800 rows of [B,T] flattened, row m = b*T + t

#define ALPHA_LO 0.81873077f
#define ALPHA_HI 0.96078944f
#define BETA_LO  0.96721610f
#define BETA_HI  0.99169440f

#define KC1 100              // K-chunk for input projection (700 = 7*100)
#define KC2 128              // K-chunk for recurrence      (1024 = 8*128)

// ---------------------------------------------------------------------------
// Async global->LDS staging of a 16-column B panel, stored transposed
// (column-major: buf[n*KC + k]) so each lane's K-pair is one aligned b64 read.
// Uses the CDNA5 async-DMA path (ASYNCcnt) instead of load+ds_store.
__device__ __forceinline__ void stage_panel_async(const float* __restrict__ G,
                                                  int k0, int col0, int ldg,
                                                  float* buf, int KC, int tid) {
  for (int i = tid; i < KC * 16; i += 128) {
    int r = i >> 4, cc = i & 15;
    // low 32 bits of a generic pointer to LDS == LDS byte offset
    unsigned lds_off = (unsigned)(unsigned long long)&buf[cc * KC + r];
    const float* g = &G[(size_t)(k0 + r) * ldg + (col0 + cc)];
    asm volatile("global_load_async_to_lds_b32 %0, %1, off"
                 :: "v"(lds_off), "v"(g) : "memory");
  }
}

__device__ __forceinline__ void wait_async0() {
  asm volatile("s_wait_asynccnt 0x0" ::: "memory");
}

// ---------------------------------------------------------------------------
// Prep: clip per-neuron params
__global__ void prep_params(const float* __restrict__ alpha, const float* __restrict__ beta,
                            const float* __restrict__ a, const float* __restrict__ b,
                            float* __restrict__ alpha_c, float* __restrict__ beta_c,
                            float* __restrict__ a_c, float* __restrict__ b_c) {
  int h = blockIdx.x * blockDim.x + threadIdx.x;
  if (h >= H_) return;
  alpha_c[h] = fminf(fmaxf(alpha[h], ALPHA_LO), ALPHA_HI);
  beta_c[h]  = fminf(fmaxf(beta[h],  BETA_LO),  BETA_HI);
  a_c[h]     = fminf(fmaxf(a[h], -1.0f), 1.0f);
  b_c[h]     = fminf(fmaxf(b[h],  0.0f), 2.0f);
}

// Prep: copy recurrent state (state updated in place across the 200 steps)
__global__ void prep_state(const float* __restrict__ ut0, const float* __restrict__ wt0,
                           float* __restrict__ ut, float* __restrict__ wt) {
  int i = blockIdx.x * blockDim.x + threadIdx.x;
  if (i >= B_ * H_) return;
  ut[i] = ut0[i];
  wt[i] = wt0[i];
}

// Prep: V0 = V_w with zeroed diagonal
__global__ void prep_v(const float* __restrict__ V_w, float* __restrict__ V0) {
  int i = blockIdx.x * blockDim.x + threadIdx.x;
  if (i >= H_ * H_) return;
  int row = i / H_, col = i - row * H_;
  V0[i] = (row == col) ? 0.0f : V_w[i];
}

// ---------------------------------------------------------------------------
// GEMM1: Wx[m,h] = sum_k x[m,k] * W_w[k,h] + W_b[h]
// Block = 4 waves; each wave owns one 16-row M-tile, all share the 16-col
// B panel, double-buffered in LDS via async global->LDS loads.
__global__ __launch_bounds__(128)
void gemm_wx(const float* __restrict__ x, const float* __restrict__ W,
             const float* __restrict__ Wb, float* __restrict__ Wx) {
  __shared__ __align__(16) float Bs[2][16 * KC1];
  const int tid    = threadIdx.x;
  const int lane   = tid & 31;
  const int wave   = tid >> 5;                        // 0..3 -> M sub-tile
  const int tile_n = blockIdx.x;                      // H/16 tiles
  const int m0     = blockIdx.y * 64 + wave * 16;
  const int nl  = lane & 15;
  const int kg  = (lane >> 4) * 2;                    // 0 or 2
  const int n   = tile_n * 16 + nl;
  const float* xrow = x + (size_t)(m0 + nl) * D_;
  const int nchunks = D_ / KC1;

  stage_panel_async(W, 0, tile_n * 16, H_, Bs[0], KC1, tid);
  wait_async0();
  __syncthreads();

  v8f c = {};
  for (int chunk = 0; chunk < nchunks; ++chunk) {
    const int k0 = chunk * KC1;
    if (chunk + 1 < nchunks)
      stage_panel_async(W, k0 + KC1, tile_n * 16, H_, Bs[(chunk + 1) & 1], KC1, tid);
    const float* Bsc = Bs[chunk & 1];
#pragma unroll 5
    for (int kk = 0; kk < KC1; kk += 4) {
      v2f av, bv;
      av.x = xrow[k0 + kk + kg];
      av.y = xrow[k0 + kk + kg + 1];
      const float2 bp = *(const float2*)&Bsc[nl * KC1 + kk + kg];
      bv.x = bp.x; bv.y = bp.y;
      c = __builtin_amdgcn_wmma_f32_16x16x4_f32(false, av, false, bv,
                                                (short)0, c, false, false);
    }
    wait_async0();      // own async ops for next panel done
    __syncthreads();    // whole block done reading cur + staging next
  }
  const float bias = Wb[n];
  const int rbase = m0 + 8 * (lane >> 4);
#pragma unroll
  for (int r = 0; r < 8; ++r)
    Wx[(size_t)(rbase + r) * H_ + n] = c[r] + bias;
}

// ---------------------------------------------------------------------------
// One RadLIF timestep. Grid = H/16 blocks; block = 4 waves covering all 64
// batch rows of one 16-col state tile, sharing a double-buffered V0 panel.
//   wt' = beta*wt + a*ut + b*s_prev
//   ut' = alpha*(ut - s_prev) + (1-alpha)*(Wx[:,t,:] + s_prev@V0 - wt')
//   s_t = (ut' - 1 > 0)
__global__ __launch_bounds__(128)
void radlif_step(const float* __restrict__ s_prev, long sp_stride,
                 const float* __restrict__ V0, const float* __restrict__ Wx,
                 float* __restrict__ ut, float* __restrict__ wt,
                 const float* __restrict__ alpha_c, const float* __restrict__ beta_c,
                 const float* __restrict__ a_c, const float* __restrict__ b_c,
                 float* __restrict__ s_out, int t) {
  __shared__ __align__(16) float Bs[2][16 * KC2];
  const int tid    = threadIdx.x;
  const int lane   = tid & 31;
  const int wave   = tid >> 5;                        // 0..3 -> batch sub-tile
  const int tile_n = blockIdx.x;                      // H/16 tiles
  const int m0     = wave * 16;                       // batch rows of this wave
  const int nl  = lane & 15;
  const int kg  = (lane >> 4) * 2;
  const int h   = tile_n * 16 + nl;
  const float* sprow = s_prev + (size_t)(m0 + nl) * (size_t)sp_stride;
  const int nchunks = H_ / KC2;

  stage_panel_async(V0, 0, tile_n * 16, H_, Bs[0], KC2, tid);
  wait_async0();
  __syncthreads();

  v8f c = {};
  for (int chunk = 0; chunk < nchunks; ++chunk) {
    const int k0 = chunk * KC2;
    if (chunk + 1 < nchunks)
      stage_panel_async(V0, k0 + KC2, tile_n * 16, H_, Bs[(chunk + 1) & 1], KC2, tid);
    const float* Bsc = Bs[chunk & 1];
#pragma unroll 8
    for (int kk = 0; kk < KC2; kk += 4) {
      v2f av, bv;
      av.x = sprow[k0 + kk + kg];
      av.y = sprow[k0 + kk + kg + 1];
      const float2 bp = *(const float2*)&Bsc[nl * KC2 + kk + kg];
      bv.x = bp.x; bv.y = bp.y;
      c = __builtin_amdgcn_wmma_f32_16x16x4_f32(false, av, false, bv,
                                                (short)0, c, false, false);
    }
    wait_async0();
    __syncthreads();
  }

  const float al = alpha_c[h];
  const float be = beta_c[h];
  const float aa = a_c[h];
  const float bb = b_c[h];
  const int rbase = m0 + 8 * (lane >> 4);
#pragma unroll
  for (int r = 0; r < 8; ++r) {
    const int m = rbase + r;                       // batch index 0..63
    const size_t sidx = (size_t)m * H_ + h;        // state index
    const float u  = ut[sidx];
    const float w  = wt[sidx];
    const float sp = s_prev[(size_t)m * (size_t)sp_stride + h];
    const float wn = be * w + aa * u + bb * sp;
    const float wx = Wx[((size_t)m * T_ + t) * H_ + h];
    const float un = al * (u - sp) + (1.0f - al) * (wx + c[r] - wn);
    const float sn = (un - 1.0f) > 0.0f ? 1.0f : 0.0f;
    ut[sidx] = un;
    wt[sidx] = wn;
    s_out[((size_t)m * T_ + t) * H_ + h] = sn;
  }
}

// ---------------------------------------------------------------------------
// Readout projection: Wxr[m,o] = s[m,:] . Wr_w[:,o] + Wr_b[o]   (O=20, tiny)
__global__ void readout_gemm(const float* __restrict__ s, const float* __restrict__ Wr_w,
                             const float* __restrict__ Wr_b, float* __restrict__ Wxr) {
  int id = blockIdx.x * blockDim.x + threadIdx.x;
  if (id >= M_ * O_) return;
  int m = id / O_;
  int o = id - m * O_;
  const float* srow = s + (size_t)m * H_;
  float acc = 0.0f;
  for (int k = 0; k < H_; ++k) acc += srow[k] * Wr_w[(size_t)k * O_ + o];
  Wxr[id] = acc + Wr_b[o];
}

// Sequential leaky readout + softmax accumulation; one thread per batch.
__global__ void readout_scan(const float* __restrict__ Wxr,
                             const float* __restrict__ alpha_read,
                             const float* __restrict__ ur0,
                             float* __restrict__ out) {
  int bidx = blockIdx.x * blockDim.x + threadIdx.x;
  if (bidx >= B_) return;
  float u[O_], acc[O_], ar_raw[O_], gain[O_];
#pragma unroll
  for (int o = 0; o < O_; ++o) {
    float ar = alpha_read[o];
    ar_raw[o] = ar;                                          // unclamped decay
    gain[o] = 1.0f - fminf(fmaxf(ar, ALPHA_LO), ALPHA_HI);   // clamped gain
    u[o]   = ur0[bidx * O_ + o];
    acc[o] = 0.0f;
  }
  for (int t = 0; t < T_; ++t) {
    const float* wx = Wxr + ((size_t)bidx * T_ + t) * O_;
    float mx = -3.402823466e+38f;
#pragma unroll
    for (int o = 0; o < O_; ++o) {
      u[o] = ar_raw[o] * u[o] + gain[o] * wx[o];
      mx = fmaxf(mx, u[o]);
    }
    float e[O_], se = 0.0f;
#pragma unroll
    for (int o = 0; o < O_; ++o) { e[o] = expf(u[o] - mx); se += e[o]; }
    float inv = 1.0f / se;
#pragma unroll
    for (int o = 0; o < O_; ++o) acc[o] += e[o] * inv;
  }
#pragma unroll
  for (int o = 0; o < O_; ++o) out[bidx * O_ + o] = acc[o];
}

// Firing rates: mean over (b,t) per h; coalesced across h.
__global__ void firing_rates(const float* __restrict__ s, float* __restrict__ fr) {
  int h = blockIdx.x * blockDim.x + threadIdx.x;
  if (h >= H_) return;
  float acc = 0.0f;
  for (int m = 0; m < M_; ++m) acc += s[(size_t)m * H_ + h];
  fr[h] = acc * (1.0f / (float)M_);
}

// ---------------------------------------------------------------------------
extern "C" void kernel_launch(void* const* d_in, const int* in_sizes, int n_in,
                              void* d_out, int out_size, void* d_ws, size_t ws_size,
                              hipStream_t stream) {
  (void)in_sizes; (void)n_in; (void)out_size; (void)ws_size;
  const float* x          = (const float*)d_in[0];
  // d_in[1] = mask (all-ones; unused by reference math)
  const float* W_w        = (const float*)d_in[2];
  const float* W_b        = (const float*)d_in[3];
  const float* V_w        = (const float*)d_in[4];
  const float* alpha      = (const float*)d_in[5];
  const float* beta       = (const float*)d_in[6];
  const float* a          = (const float*)d_in[7];
  const float* b          = (const float*)d_in[8];
  const float* Wr_w       = (const float*)d_in[9];
  const float* Wr_b       = (const float*)d_in[10];
  const float* alpha_read = (const float*)d_in[11];
  const float* ut0        = (const float*)d_in[12];
  const float* wt0        = (const float*)d_in[13];
  const float* st0        = (const float*)d_in[14];
  const float* ur0        = (const float*)d_in[15];

  float* out = (float*)d_out;          // [B,O]
  float* fr  = out + B_ * O_;          // [H]
  float* s   = fr + H_;                // [B,T,H]

  float* ws      = (float*)d_ws;
  float* Wx      = ws;  ws += (size_t)M_ * H_;     // 13,107,200
  float* ut      = ws;  ws += (size_t)B_ * H_;
  float* wt      = ws;  ws += (size_t)B_ * H_;
  float* V0      = ws;  ws += (size_t)H_ * H_;
  float* alpha_c = ws;  ws += H_;
  float* beta_c  = ws;  ws += H_;
  float* a_c     = ws;  ws += H_;
  float* b_c     = ws;  ws += H_;
  float* Wxr     = ws;  ws += (size_t)M_ * O_;

  prep_params<<<(H_ + 255) / 256, 256, 0, stream>>>(alpha, beta, a, b,
                                                    alpha_c, beta_c, a_c, b_c);
  prep_state<<<(B_ * H_ + 255) / 256, 256, 0, stream>>>(ut0, wt0, ut, wt);
  prep_v<<<(H_ * H_ + 255) / 256, 256, 0, stream>>>(V_w, V0);

  gemm_wx<<<dim3(H_ / 16, M_ / 64), 128, 0, stream>>>(x, W_w, W_b, Wx);

  for (int t = 0; t < T_; ++t) {
    const float* sp = (t == 0) ? st0 : (s + (size_t)(t - 1) * H_);
    long sp_stride  = (t == 0) ? (long)H_ : (long)T_ * H_;
    radlif_step<<<H_ / 16, 128, 0, stream>>>(
        sp, sp_stride, V0, Wx, ut, wt, alpha_c, beta_c, a_c, b_c, s, t);
  }

  readout_gemm<<<(M_ * O_ + 255) / 256, 256, 0, stream>>>(s, Wr_w, Wr_b, Wxr);
  readout_scan<<<1, 64, 0, stream>>>(Wxr, alpha_read, ur0, out);
  firing_rates<<<(H_ + 255) / 256, 256, 0, stream>>>(s, fr);
}